// share_Encoder_12841952215154
// MI455X (gfx1250) — compile-verified
//
#include <hip/hip_runtime.h>

#define U_CNT 100000
#define I_CNT 50000
#define N_CNT 150000
#define D_DIM 64
#define NNZ_CNT 6000000
#define L_LAYERS 3

typedef __attribute__((ext_vector_type(2))) float v2f;
typedef __attribute__((ext_vector_type(8))) float v8f;

// ---------------------------------------------------------------------------
// Init: ego0 = concat(user_emb, item_emb); acc = 0
// ---------------------------------------------------------------------------
__global__ __launch_bounds__(256) void init_ego_acc(
    const float* __restrict__ user, const float* __restrict__ item,
    float* __restrict__ ego, float* __restrict__ acc) {
  long long i = (long long)blockIdx.x * blockDim.x + threadIdx.x;
  const long long total = (long long)N_CNT * D_DIM;
  if (i >= total) return;
  const long long usz = (long long)U_CNT * D_DIM;
  ego[i] = (i < usz) ? user[i] : item[i - usz];
  acc[i] = 0.0f;
}

__global__ __launch_bounds__(256) void zero_buf(float* __restrict__ p, long long n) {
  long long i = (long long)blockIdx.x * blockDim.x + threadIdx.x;
  if (i < n) p[i] = 0.0f;
}

__global__ __launch_bounds__(256) void acc_add(float* __restrict__ acc,
                                               const float* __restrict__ cur,
                                               long long n) {
  long long i = (long long)blockIdx.x * blockDim.x + threadIdx.x;
  if (i < n) acc[i] += cur[i];
}

// ---------------------------------------------------------------------------
// SpMM: out[row] += val * x[col], edge-parallel, 16 threads per edge.
// Lane group of 16 reads x[col] as coalesced float4s (256B row, L2-resident),
// scatters with native f32 atomics (row table fits in the 192MB L2).
// ---------------------------------------------------------------------------
__global__ __launch_bounds__(256) void spmm_edges(
    const int* __restrict__ row, const int* __restrict__ col,
    const float* __restrict__ val, const float* __restrict__ x,
    float* __restrict__ out) {
  long long g = (long long)blockIdx.x * blockDim.x + threadIdx.x;
  long long e = g >> 4;
  int part = (int)(g & 15);
  if (e >= (long long)NNZ_CNT) return;
  int r = row[e];
  int c = col[e];
  float v = val[e];
  const float4* xs = (const float4*)(x + (long long)c * D_DIM);
  float4 d = xs[part];
  float* o = out + (long long)r * D_DIM + part * 4;
  unsafeAtomicAdd(o + 0, v * d.x);
  unsafeAtomicAdd(o + 1, v * d.y);
  unsafeAtomicAdd(o + 2, v * d.z);
  unsafeAtomicAdd(o + 3, v * d.w);
}

// ---------------------------------------------------------------------------
// Fused 2-layer MLP: out = relu(x*scale @ W1^T + b1) @ W2^T + b2
// One wave per 16-row tile, f32 WMMA 16x16x4, relu'd layer-1 tile staged in
// padded LDS (row stride 66 to dodge the 64-bank wrap) to re-marshal
// C-layout -> A-layout for layer 2.
//
// f32 WMMA fragment layouts (ISA 7.12.2):
//   A 16x4 : lane m = lane&15 is matrix row; VGPR0/1 hold K = 2*(lane>>4)+{0,1}
//   B 4x16 : lane m is matrix col N; VGPR0/1 hold K = 2*(lane>>4)+{0,1}
//   C/D    : VGPR j holds row M = j + 8*(lane>>4), col N = lane&15
// ---------------------------------------------------------------------------
__global__ __launch_bounds__(256) void mlp2(
    const float* __restrict__ x, int nrows,
    const float* __restrict__ W1, const float* __restrict__ b1,
    const float* __restrict__ W2, const float* __restrict__ b2,
    float* __restrict__ out, float scale) {
  __shared__ float lds[8][16][66];

  int wave = threadIdx.x >> 5;
  int lane = threadIdx.x & 31;
  int tile = blockIdx.x * 8 + wave;
  if (tile * 16 >= nrows) return;  // wave-uniform: EXEC stays all-ones for WMMA

  const int m = lane & 15;   // A row / B col / D col
  const int kh = lane >> 4;  // K half: 0 -> K{0,1}, 1 -> K{2,3}
  float(*h)[66] = lds[wave];

  const float* xrow = x + (long long)tile * 16 * D_DIM;

  // ----- Layer 1: h = relu(x*scale @ W1^T + b1) -----
#pragma unroll
  for (int n0 = 0; n0 < 64; n0 += 16) {
    v8f c = {};
#pragma unroll
    for (int k0 = 0; k0 < 64; k0 += 4) {
      int ka = k0 + 2 * kh;
      v2f a;
      a.x = xrow[m * 64 + ka] * scale;
      a.y = xrow[m * 64 + ka + 1] * scale;
      v2f bm;  // B[k][n] = W1[n][k]
      bm.x = W1[(n0 + m) * 64 + ka];
      bm.y = W1[(n0 + m) * 64 + ka + 1];
      c = __builtin_amdgcn_wmma_f32_16x16x4_f32(
          false, a, false, bm, (short)0, c, false, false);
    }
    float bias = b1[n0 + m];
#pragma unroll
    for (int j = 0; j < 8; ++j) {
      float vv = c[j] + bias;
      vv = vv > 0.0f ? vv : 0.0f;
      h[j + 8 * kh][n0 + m] = vv;  // store D tile: row j+8*kh, col n0+m
    }
  }
  // Same-wave LDS RAW: compiler inserts s_wait_dscnt; no cross-wave sharing.

  // ----- Layer 2: out = h @ W2^T + b2 -----
  float* orow = out + (long long)tile * 16 * D_DIM;
#pragma unroll
  for (int n0 = 0; n0 < 64; n0 += 16) {
    v8f c = {};
#pragma unroll
    for (int k0 = 0; k0 < 64; k0 += 4) {
      int ka = k0 + 2 * kh;
      v2f a;
      a.x = h[m][ka];
      a.y = h[m][ka + 1];
      v2f bm;
      bm.x = W2[(n0 + m) * 64 + ka];
      bm.y = W2[(n0 + m) * 64 + ka + 1];
      c = __builtin_amdgcn_wmma_f32_16x16x4_f32(
          false, a, false, bm, (short)0, c, false, false);
    }
    float bias = b2[n0 + m];
#pragma unroll
    for (int j = 0; j < 8; ++j) {
      orow[(j + 8 * kh) * 64 + n0 + m] = c[j] + bias;
    }
  }
}

// ---------------------------------------------------------------------------
// Launch
// ---------------------------------------------------------------------------
extern "C" void kernel_launch(void* const* d_in, const int* in_sizes, int n_in,
                              void* d_out, int out_size, void* d_ws, size_t ws_size,
                              hipStream_t stream) {
  const float* user_emb = (const float*)d_in[0];
  const float* item_emb = (const float*)d_in[1];
  const int* edge_row   = (const int*)d_in[2];
  const int* edge_col   = (const int*)d_in[3];
  const float* edge_val = (const float*)d_in[4];
  const float* W_sg1 = (const float*)d_in[5];
  const float* b_sg1 = (const float*)d_in[6];
  const float* W_sg2 = (const float*)d_in[7];
  const float* b_sg2 = (const float*)d_in[8];
  const float* W_uf1 = (const float*)d_in[9];
  const float* b_uf1 = (const float*)d_in[10];
  const float* W_uf2 = (const float*)d_in[11];
  const float* b_uf2 = (const float*)d_in[12];
  const float* W_if1 = (const float*)d_in[13];
  const float* b_if1 = (const float*)d_in[14];
  const float* W_if2 = (const float*)d_in[15];
  const float* b_if2 = (const float*)d_in[16];
  float* out = (float*)d_out;

  const long long NV = (long long)N_CNT * D_DIM;  // 9.6M floats
  float* bufA = (float*)d_ws;
  float* bufB = bufA + NV;
  float* acc  = bufB + NV;

  dim3 blk(256);
  int gridNV = (int)((NV + 255) / 256);

  hipLaunchKernelGGL(init_ego_acc, dim3(gridNV), blk, 0, stream,
                     user_emb, item_emb, bufA, acc);

  float* cur = bufA;
  float* nxt = bufB;
  long long spmm_threads = (long long)NNZ_CNT * 16;
  int spmm_grid = (int)((spmm_threads + 255) / 256);
  for (int l = 0; l < L_LAYERS; ++l) {
    hipLaunchKernelGGL(zero_buf, dim3(gridNV), blk, 0, stream, nxt, NV);
    hipLaunchKernelGGL(spmm_edges, dim3(spmm_grid), blk, 0, stream,
                       edge_row, edge_col, edge_val, cur, nxt);
    hipLaunchKernelGGL(acc_add, dim3(gridNV), blk, 0, stream, acc, nxt, NV);
    float* t = cur; cur = nxt; nxt = t;
  }

  const float scale = 1.0f / (float)L_LAYERS;

  // h_shared: out[:U] ++ out[U:] is simply the full contiguous N x 64 block.
  int gs = (N_CNT / 16 + 7) / 8;
  hipLaunchKernelGGL(mlp2, dim3(gs), blk, 0, stream,
                     acc, N_CNT, W_sg1, b_sg1, W_sg2, b_sg2, out, scale);
  // h_user
  int gu = (U_CNT / 16 + 7) / 8;
  hipLaunchKernelGGL(mlp2, dim3(gu), blk, 0, stream,
                     acc, U_CNT, W_uf1, b_uf1, W_uf2, b_uf2, out + NV, scale);
  // h_item
  int gi = (I_CNT / 16 + 7) / 8;
  hipLaunchKernelGGL(mlp2, dim3(gi), blk, 0, stream,
                     acc + (long long)U_CNT * D_DIM, I_CNT,
                     W_if1, b_if1, W_if2, b_if2,
                     out + NV + (long long)U_CNT * D_DIM, scale);
}